// lGNN_22960895164798
// MI455X (gfx1250) — compile-verified
//
#include <hip/hip_runtime.h>
#include <hip/hip_bf16.h>

typedef __attribute__((ext_vector_type(2))) float v2f;
typedef __attribute__((ext_vector_type(8))) float v8f;

#define N_NODES 50000
#define N_EDGES 800000
#define N_GRAPHS 256
#define IN_DIM 6
#define C 96
#define C3 288
#define ROW_TILES (N_NODES / 16)   // 3125 exactly

// ---------------------------------------------------------------------------
// Utility: zero fill
// ---------------------------------------------------------------------------
__global__ void zero_f32(float* __restrict__ p, int n) {
    int i = blockIdx.x * blockDim.x + threadIdx.x;
    if (i < n) p[i] = 0.0f;
}
__global__ void zero_u32(unsigned* __restrict__ p, int n) {
    int i = blockIdx.x * blockDim.x + threadIdx.x;
    if (i < n) p[i] = 0u;
}

// ---------------------------------------------------------------------------
// Input transform: h = x @ W_in + b_in      [N,6]@[6,96] -> [N,96]
// ---------------------------------------------------------------------------
__global__ void input_transform(const float* __restrict__ x,
                                const float* __restrict__ Win,
                                const float* __restrict__ bin,
                                float* __restrict__ h) {
    int idx = blockIdx.x * blockDim.x + threadIdx.x;
    if (idx >= N_NODES * C) return;
    int n = idx / C, c = idx % C;
    float acc = bin[c];
#pragma unroll
    for (int k = 0; k < IN_DIM; ++k)
        acc += x[n * IN_DIM + k] * Win[k * C + c];
    h[idx] = acc;
}

// ---------------------------------------------------------------------------
// GEMM (NN): Out[N,96] = A[N,96] @ W[96,96]  (W row-major), fp32 WMMA 16x16x4
// One 16x16 output tile per wave; W staged in LDS.
// ---------------------------------------------------------------------------
__global__ void gemm_nn96(const float* __restrict__ A,
                          const float* __restrict__ W,
                          float* __restrict__ Out) {
    __shared__ float Ws[C * C];
    for (int i = threadIdx.x; i < C * C; i += blockDim.x) Ws[i] = W[i];
    __syncthreads();

    const int wave = threadIdx.x >> 5;
    const int lane = threadIdx.x & 31;
    const int half = lane >> 4;       // 0: K pair {0,1}, 1: K pair {2,3}
    const int lid  = lane & 15;

    const int nTiles = ROW_TILES * (C / 16);  // 3125 * 6
    int tile = blockIdx.x * 8 + wave;
    if (tile >= nTiles) return;
    const int rt = tile / (C / 16);
    const int ct = tile % (C / 16);
    const int r0 = rt * 16, c0 = ct * 16;

    v8f acc = {};
    const float* arow = A + (r0 + lid) * C + 2 * half;  // A frag: M=lid, K=k+2*half+{0,1}
#pragma unroll
    for (int k = 0; k < C; k += 4) {
        v2f a = *(const v2f*)(arow + k);
        v2f b;
        b.x = Ws[(k + 2 * half) * C + c0 + lid];        // B frag: K=k+2*half, N=lid
        b.y = Ws[(k + 2 * half + 1) * C + c0 + lid];
        acc = __builtin_amdgcn_wmma_f32_16x16x4_f32(false, a, false, b,
                                                    (short)0, acc, false, false);
    }
    float* orow = Out + c0 + lid;
#pragma unroll
    for (int r = 0; r < 8; ++r)
        orow[(r0 + r + 8 * half) * C] = acc[r];        // D: M=r+8*half, N=lid
}

// ---------------------------------------------------------------------------
// GEMM (NT) + bias: Out[N,288] = A[N,96] @ Wt^T + bias, Wt is [288,96] row-major
// B fragment is contiguous float2 from Wt (L2 resident, 110KB).
// ---------------------------------------------------------------------------
__global__ void gemm_nt288(const float* __restrict__ A,
                           const float* __restrict__ Wt,
                           const float* __restrict__ bias,
                           float* __restrict__ Out) {
    const int wave = threadIdx.x >> 5;
    const int lane = threadIdx.x & 31;
    const int half = lane >> 4;
    const int lid  = lane & 15;

    const int nTiles = ROW_TILES * (C3 / 16);  // 3125 * 18
    int tile = blockIdx.x * 8 + wave;
    if (tile >= nTiles) return;
    const int rt = tile / (C3 / 16);
    const int ct = tile % (C3 / 16);
    const int r0 = rt * 16, c0 = ct * 16;

    v8f acc = {};
    const float* arow = A + (r0 + lid) * C + 2 * half;
    const float* brow = Wt + (c0 + lid) * C + 2 * half;  // B[K][N] = Wt[N][K]
#pragma unroll
    for (int k = 0; k < C; k += 4) {
        v2f a = *(const v2f*)(arow + k);
        v2f b = *(const v2f*)(brow + k);
        acc = __builtin_amdgcn_wmma_f32_16x16x4_f32(false, a, false, b,
                                                    (short)0, acc, false, false);
    }
    const float bv = bias[c0 + lid];
    float* orow = Out + c0 + lid;
#pragma unroll
    for (int r = 0; r < 8; ++r)
        orow[(r0 + r + 8 * half) * C3] = acc[r] + bv;
}

// ---------------------------------------------------------------------------
// Edge scatter-add: agg[dst] += m[src]   (4 channels / thread, float4 load)
// ---------------------------------------------------------------------------
__global__ void scatter_edges(const float* __restrict__ m,
                              const int* __restrict__ ei,
                              float* __restrict__ agg) {
    int t = blockIdx.x * blockDim.x + threadIdx.x;
    const int total = N_EDGES * (C / 4);
    if (t >= total) return;
    int e  = t / (C / 4);
    int c4 = (t % (C / 4)) * 4;
    int src = ei[e];
    int dst = ei[N_EDGES + e];
    float4 v = *(const float4*)(m + src * C + c4);
    float* ap = agg + dst * C + c4;
    atomicAdd(ap + 0, v.x);
    atomicAdd(ap + 1, v.y);
    atomicAdd(ap + 2, v.z);
    atomicAdd(ap + 3, v.w);
}

// ---------------------------------------------------------------------------
// GRU elementwise: h = (1-z)*n + z*h   (gate order r,z,n; in-place on h)
// ---------------------------------------------------------------------------
__global__ void gru_elem(const float* __restrict__ gi,
                         const float* __restrict__ gh,
                         float* __restrict__ h) {
    int idx = blockIdx.x * blockDim.x + threadIdx.x;
    if (idx >= N_NODES * C) return;
    int n = idx / C, c = idx % C;
    float ir = gi[n * C3 + c],        hr = gh[n * C3 + c];
    float iz = gi[n * C3 + C + c],    hz = gh[n * C3 + C + c];
    float in_ = gi[n * C3 + 2*C + c], hn = gh[n * C3 + 2*C + c];
    float r = 1.0f / (1.0f + __expf(-(ir + hr)));
    float z = 1.0f / (1.0f + __expf(-(iz + hz)));
    float nc = tanhf(in_ + r * hn);
    float ho = h[idx];
    h[idx] = (1.0f - z) * nc + z * ho;
}

// ---------------------------------------------------------------------------
// Attentional aggregation: gate scores + order-preserving-uint segment max
// ---------------------------------------------------------------------------
__device__ __forceinline__ unsigned enc_f32(float f) {
    unsigned b = __float_as_uint(f);
    return (b & 0x80000000u) ? ~b : (b | 0x80000000u);
}
__device__ __forceinline__ float dec_f32(unsigned u) {
    unsigned b = (u & 0x80000000u) ? (u & 0x7fffffffu) : ~u;
    return __uint_as_float(b);
}

__global__ void gate_kernel(const float* __restrict__ h,
                            const float* __restrict__ Wg,
                            const float* __restrict__ bg,
                            const int* __restrict__ batch,
                            float* __restrict__ gate,
                            unsigned* __restrict__ gmax) {
    int i = blockIdx.x * blockDim.x + threadIdx.x;
    if (i >= N_NODES) return;
    float acc = bg[0];
#pragma unroll 8
    for (int k = 0; k < C; ++k) acc += h[i * C + k] * Wg[k];
    gate[i] = acc;
    atomicMax(&gmax[batch[i]], enc_f32(acc));
}

__global__ void exp_kernel(float* __restrict__ gate_e,
                           const int* __restrict__ batch,
                           const unsigned* __restrict__ gmax,
                           float* __restrict__ denom) {
    int i = blockIdx.x * blockDim.x + threadIdx.x;
    if (i >= N_NODES) return;
    int g = batch[i];
    unsigned u = gmax[g];
    float gm = (u == 0u) ? 0.0f : dec_f32(u);   // empty segment -> 0 (matches ref)
    float ev = __expf(gate_e[i] - gm);
    gate_e[i] = ev;
    atomicAdd(&denom[g], ev);
}

__global__ void pool_kernel(const float* __restrict__ h,
                            const float* __restrict__ e,
                            const int* __restrict__ batch,
                            const float* __restrict__ denom,
                            float* __restrict__ pooled) {
    int idx = blockIdx.x * blockDim.x + threadIdx.x;
    if (idx >= N_NODES * C) return;
    int n = idx / C, c = idx % C;
    int g = batch[n];
    float alpha = e[n] / (denom[g] + 1e-16f);
    atomicAdd(&pooled[g * C + c], alpha * h[idx]);
}

// ---------------------------------------------------------------------------
// MLP head: relu(relu(pooled@W1+b1)@W2+b2)@W_out+b_out ; one block per graph
// ---------------------------------------------------------------------------
__global__ void head_kernel(const float* __restrict__ pooled,
                            const float* __restrict__ W1, const float* __restrict__ b1,
                            const float* __restrict__ W2, const float* __restrict__ b2,
                            const float* __restrict__ Wo, const float* __restrict__ bo,
                            float* __restrict__ out) {
    __shared__ float s[C];
    __shared__ float t[C];
    const int g = blockIdx.x;
    const int c = threadIdx.x;   // 96 threads
    s[c] = pooled[g * C + c];
    __syncthreads();
    float a1 = b1[c];
#pragma unroll 8
    for (int k = 0; k < C; ++k) a1 += s[k] * W1[k * C + c];
    t[c] = fmaxf(a1, 0.0f);
    __syncthreads();
    float a2 = b2[c];
#pragma unroll 8
    for (int k = 0; k < C; ++k) a2 += t[k] * W2[k * C + c];
    s[c] = fmaxf(a2, 0.0f) * Wo[c];
    __syncthreads();
    if (c == 0) {
        float sum = bo[0];
        for (int k = 0; k < C; ++k) sum += s[k];
        out[g] = sum;
    }
}

// ---------------------------------------------------------------------------
// Host-side launcher
// ---------------------------------------------------------------------------
extern "C" void kernel_launch(void* const* d_in, const int* in_sizes, int n_in,
                              void* d_out, int out_size, void* d_ws, size_t ws_size,
                              hipStream_t stream) {
    const float* x      = (const float*)d_in[0];
    const int*   ei     = (const int*)  d_in[1];
    const int*   batch  = (const int*)  d_in[2];
    const float* W_in   = (const float*)d_in[3];
    const float* b_in   = (const float*)d_in[4];
    const float* conv_w = (const float*)d_in[5];
    const float* w_ih   = (const float*)d_in[6];
    const float* w_hh   = (const float*)d_in[7];
    const float* b_ih   = (const float*)d_in[8];
    const float* b_hh   = (const float*)d_in[9];
    const float* W_gate = (const float*)d_in[10];
    const float* b_gate = (const float*)d_in[11];
    const float* W_fc1  = (const float*)d_in[12];
    const float* b_fc1  = (const float*)d_in[13];
    const float* W_fc2  = (const float*)d_in[14];
    const float* b_fc2  = (const float*)d_in[15];
    const float* W_out  = (const float*)d_in[16];
    const float* b_out  = (const float*)d_in[17];
    float* out = (float*)d_out;

    // workspace layout (~173 MB)
    float* ws = (float*)d_ws;
    float* h      = ws;  ws += (size_t)N_NODES * C;
    float* m      = ws;  ws += (size_t)N_NODES * C;
    float* agg    = ws;  ws += (size_t)N_NODES * C;
    float* gi     = ws;  ws += (size_t)N_NODES * C3;
    float* gh     = ws;  ws += (size_t)N_NODES * C3;
    float* gate_e = ws;  ws += N_NODES;
    float* denom  = ws;  ws += N_GRAPHS;
    float* pooled = ws;  ws += N_GRAPHS * C;
    unsigned* gmax = (unsigned*)ws;

    const int NC   = N_NODES * C;            // 4.8M
    const int T256 = 256;
    const int nnBlocks = (ROW_TILES * (C / 16) + 7) / 8;    // 6 col tiles, 8 waves/blk
    const int ntBlocks = (ROW_TILES * (C3 / 16) + 7) / 8;   // 18 col tiles

    // h = x @ W_in + b_in
    input_transform<<<(NC + T256 - 1) / T256, T256, 0, stream>>>(x, W_in, b_in, h);

    // 3 GatedGraphConv layers
    for (int l = 0; l < 3; ++l) {
        gemm_nn96<<<nnBlocks, 256, 0, stream>>>(h, conv_w + (size_t)l * C * C, m);
        zero_f32<<<(NC + T256 - 1) / T256, T256, 0, stream>>>(agg, NC);
        scatter_edges<<<(N_EDGES * (C / 4) + T256 - 1) / T256, T256, 0, stream>>>(m, ei, agg);
        gemm_nt288<<<ntBlocks, 256, 0, stream>>>(agg, w_ih, b_ih, gi);
        gemm_nt288<<<ntBlocks, 256, 0, stream>>>(h,   w_hh, b_hh, gh);
        gru_elem<<<(NC + T256 - 1) / T256, T256, 0, stream>>>(gi, gh, h);
    }

    // attentional aggregation
    zero_f32<<<(N_GRAPHS * (C + 1) + T256 - 1) / T256, T256, 0, stream>>>(denom, N_GRAPHS * (C + 1)); // denom + pooled (contiguous)
    zero_u32<<<1, N_GRAPHS, 0, stream>>>(gmax, N_GRAPHS);
    gate_kernel<<<(N_NODES + T256 - 1) / T256, T256, 0, stream>>>(h, W_gate, b_gate, batch, gate_e, gmax);
    exp_kernel<<<(N_NODES + T256 - 1) / T256, T256, 0, stream>>>(gate_e, batch, gmax, denom);
    pool_kernel<<<(NC + T256 - 1) / T256, T256, 0, stream>>>(h, gate_e, batch, denom, pooled);

    // MLP head
    head_kernel<<<N_GRAPHS, C, 0, stream>>>(pooled, W_fc1, b_fc1, W_fc2, b_fc2, W_out, b_out, out);
}